// Adagnn_with_weight_9019431321742
// MI455X (gfx1250) — compile-verified
//
#include <hip/hip_runtime.h>
#include <hip/hip_bf16.h>

typedef float v2f __attribute__((ext_vector_type(2)));
typedef float v8f __attribute__((ext_vector_type(8)));

#define FDIM 256

// ---------------------------------------------------------------------------
// Kernel 1: zero the e1 accumulator (in d_ws) before atomic scatter.
// ---------------------------------------------------------------------------
__global__ void zero_f4_kernel(float4* __restrict__ p, size_t n4) {
    size_t i = (size_t)blockIdx.x * blockDim.x + threadIdx.x;
    size_t stride = (size_t)gridDim.x * blockDim.x;
    float4 z; z.x = 0.f; z.y = 0.f; z.z = 0.f; z.w = 0.f;
    for (; i < n4; i += stride) p[i] = z;
}

// ---------------------------------------------------------------------------
// Kernel 2: pack W[256][256] into k-pair-interleaved layout:
//   Wpk2[kp*256 + n] = { W[2*kp][n], W[2*kp+1][n] }   (float2, kp = 0..127)
// so a B fragment for V_WMMA_F32_16X16X4_F32 is ONE global_load_b64.
// ---------------------------------------------------------------------------
__global__ void pack_w_kernel(const float* __restrict__ W,
                              float2* __restrict__ Wpk2) {
    int idx = blockIdx.x * blockDim.x + threadIdx.x;  // 0 .. 128*256-1
    int kp = idx >> 8;
    int n  = idx & 255;
    float2 v;
    v.x = W[(size_t)(2 * kp)     * FDIM + n];
    v.y = W[(size_t)(2 * kp + 1) * FDIM + n];
    Wpk2[idx] = v;
}

// ---------------------------------------------------------------------------
// Kernel 3: SpMM scatter. 64 threads per edge; each thread handles one float4
// of the 256-wide feature row. x (102MB) and e1 (102MB) both fit in the
// 192MB L2, so gathers + f32 atomic adds resolve at L2 bandwidth.
// ---------------------------------------------------------------------------
__global__ void spmm_scatter_kernel(const float* __restrict__ x,
                                    const int* __restrict__ edge_row,
                                    const int* __restrict__ edge_col,
                                    const float* __restrict__ edge_val,
                                    float* __restrict__ e1,
                                    int E) {
    int tid = blockIdx.x * blockDim.x + threadIdx.x;
    int e  = tid >> 6;        // 64 threads per edge
    int f4 = tid & 63;        // which float4 of the 256 features
    if (e >= E) return;
    int r = edge_row[e];
    int c = edge_col[e];
    float v = edge_val[e];
    const float4* src = (const float4*)(x + (size_t)c * FDIM);
    float4 g = src[f4];
    float* dst = e1 + (size_t)r * FDIM + (size_t)f4 * 4;
    atomicAdd(dst + 0, v * g.x);
    atomicAdd(dst + 1, v * g.y);
    atomicAdd(dst + 2, v * g.z);
    atomicAdd(dst + 3, v * g.w);
}

// ---------------------------------------------------------------------------
// Kernel 4: fused  out = (x - e1*(diag+1)) @ W + bias  using f32 WMMA.
// Each wave computes a 16-row x 128-col slab: 8 accumulators of v8f (64 VGPRs)
// so the allocator never spills; a sibling wave covers the other 128 columns.
// A-fragment (16x4 f32): lane%16 = M; the 2 VGPRs hold K = 2*(lane/16)+{0,1}.
// B-fragment (4x16 f32): lane%16 = N; the 2 VGPRs hold K = 2*(lane/16)+{0,1}
//   -> one b64 load from the packed W layout.
// C/D (16x16 f32): VGPR r -> M = r + 8*(lane/16), N = lane%16.
// The 8 B fragments per k-step are staged in registers so the loads issue as
// one clause with a single wait.
// ---------------------------------------------------------------------------
__global__ __launch_bounds__(256)
void fused_gemm_wmma_kernel(const float* __restrict__ x,
                            const float* __restrict__ e1,
                            const v2f* __restrict__ Wpk2,
                            const float* __restrict__ diag,
                            const float* __restrict__ bias,
                            float* __restrict__ out,
                            int nrows) {
    __shared__ float s_scale[FDIM];   // diag + 1, shared by all 8 waves

    const int lane = threadIdx.x & 31;
    const int wave = threadIdx.x >> 5;

    // Cooperative load of the scale vector into LDS (before any wave exits).
    s_scale[threadIdx.x] = diag[threadIdx.x] + 1.0f;
    __syncthreads();

    const int wid      = blockIdx.x * 8 + wave;  // global wave id
    const int row_base = (wid >> 1) * 16;        // 16-row tile
    const int col_base = (wid & 1) * 128;        // which 128-col half
    if (row_base >= nrows) return;   // wave-uniform: EXEC stays all-1s below

    const int half = lane >> 4;      // 0 or 1
    const int l16  = lane & 15;

    v8f zero = {};
    v8f acc[8];
#pragma unroll
    for (int i = 0; i < 8; ++i) acc[i] = zero;

    const int am = row_base + l16;   // this lane's A-matrix row
    const float* __restrict__ xrow = x  + (size_t)am * FDIM;
    const float* __restrict__ erow = e1 + (size_t)am * FDIM;
    // Lane's B base: kp = kt*2 + half, n = col_base + nt*16 + l16
    //   Wpk2[kp*256 + n] -> base + kt*512 + nt*16 (v2f units)
    const v2f* __restrict__ wbase = Wpk2 + (size_t)half * FDIM + col_base + l16;

    for (int kt = 0; kt < FDIM / 4; ++kt) {
        const int k = kt * 4 + half * 2;   // even, 8B aligned

        // A fragment: e4[m][k..k+1] = x - e1*(diag+1), computed on the fly
        float2 xv = *(const float2*)(xrow + k);
        float2 ev = *(const float2*)(erow + k);
        float2 sv = *(const float2*)(s_scale + k);   // ds_load
        v2f a;
        a.x = xv.x - ev.x * sv.x;
        a.y = xv.y - ev.y * sv.y;

        // Stage the 8 B fragments (one b64 load each, single clause+wait)
        v2f bfrag[8];
        const v2f* wk = wbase + (size_t)kt * 512;
#pragma unroll
        for (int nt = 0; nt < 8; ++nt) {
            bfrag[nt] = wk[nt * 16];
        }
#pragma unroll
        for (int nt = 0; nt < 8; ++nt) {
            acc[nt] = __builtin_amdgcn_wmma_f32_16x16x4_f32(
                /*neg_a=*/false, a, /*neg_b=*/false, bfrag[nt],
                /*c_mod=*/(short)0, acc[nt],
                /*reuse_a=*/false, /*reuse_b=*/false);
        }
    }

    // Epilogue: add bias, store per C/D VGPR layout.
#pragma unroll
    for (int nt = 0; nt < 8; ++nt) {
        const int n = col_base + nt * 16 + l16;
        const float bn = bias[n];
#pragma unroll
        for (int r = 0; r < 8; ++r) {
            const int m = row_base + r + 8 * half;
            out[(size_t)m * FDIM + n] = acc[nt][r] + bn;
        }
    }
}

// ---------------------------------------------------------------------------
// Launcher
// ---------------------------------------------------------------------------
extern "C" void kernel_launch(void* const* d_in, const int* in_sizes, int n_in,
                              void* d_out, int out_size, void* d_ws, size_t ws_size,
                              hipStream_t stream) {
    const float* x        = (const float*)d_in[0];
    const int*   edge_row = (const int*)  d_in[1];
    const int*   edge_col = (const int*)  d_in[2];
    const float* edge_val = (const float*)d_in[3];
    const float* W        = (const float*)d_in[4];
    const float* diag     = (const float*)d_in[5];
    const float* bias     = (const float*)d_in[6];
    float*       out      = (float*)d_out;

    const int N = in_sizes[0] / FDIM;               // 100000
    const int E = in_sizes[1];                      // 3200000

    // Workspace layout: e1 [N*256 f32] | Wpk [128*256 float2]
    float*  e1   = (float*)d_ws;
    float2* Wpk2 = (float2*)((char*)d_ws + (size_t)N * FDIM * sizeof(float));

    // 1) zero e1
    {
        size_t n4 = (size_t)N * (FDIM / 4);
        zero_f4_kernel<<<4096, 256, 0, stream>>>((float4*)e1, n4);
    }
    // 2) pack W into k-pair interleaved layout (32768 float2 elements)
    {
        pack_w_kernel<<<128, 256, 0, stream>>>(W, Wpk2);
    }
    // 3) SpMM scatter: 64 threads/edge -> 4 edges per 256-thread block
    {
        int blocks = (E + 3) / 4;
        spmm_scatter_kernel<<<blocks, 256, 0, stream>>>(x, edge_row, edge_col,
                                                        edge_val, e1, E);
    }
    // 4) fused elementwise + GEMM with f32 WMMA:
    //    one wave = 16 rows x 128 cols; 2 waves per row tile; 8 waves/block
    {
        int tiles  = (N + 15) / 16;          // 16-row tiles
        int waves  = tiles * 2;              // x2 column halves
        int blocks = (waves + 7) / 8;
        fused_gemm_wmma_kernel<<<blocks, 256, 0, stream>>>(x, e1, (const v2f*)Wpk2,
                                                           diag, bias, out, N);
    }
}